// WLEncoder_Categorical_57638461112698
// MI455X (gfx1250) — compile-verified
//
#include <hip/hip_runtime.h>
#include <stdint.h>

// ---------------- problem constants (match reference) ----------------
#define N_NODES   100000
#define N_FEAT    64
#define N_EDGES   6400000
#define NUM_ITERS 3

#define HASH_LOG 18
#define HASH_M   (1u << HASH_LOG)          // 262144 slots for <=100000 uniques
#define EMPTY64  0xFFFFFFFFFFFFFFFFull

// argmax tile
#define AM_NODES   128                      // nodes per block
#define AM_PAD     68                       // floats per LDS row (16B-aligned rows, de-conflicted banks)

typedef __attribute__((ext_vector_type(2))) float v2f;
typedef __attribute__((ext_vector_type(8))) float v8f;

// ---------------- device helpers ----------------
__device__ __forceinline__ unsigned long long splitmix64(unsigned long long z) {
  z += 0x9E3779B97F4A7C15ull;
  z = (z ^ (z >> 30)) * 0xBF58476D1CE4E5B9ull;
  z = (z ^ (z >> 27)) * 0x94D049BB133111EBull;
  return z ^ (z >> 31);
}

typedef __attribute__((address_space(3))) void lds_void_t;
__device__ __forceinline__ unsigned int lds_addr32(void* p) {
  // generic -> LDS addrspace, then ptrtoint gives the 32-bit LDS byte offset
  return (unsigned int)(unsigned long long)(lds_void_t*)p;
}

__device__ __forceinline__ void async_wait0() {
#if defined(__has_builtin)
#if __has_builtin(__builtin_amdgcn_s_wait_asynccnt)
  __builtin_amdgcn_s_wait_asynccnt(0);
  return;
#endif
#endif
  asm volatile("s_wait_asynccnt 0" ::: "memory");
}

// ---------------- kernels ----------------

// initial colors = argmax over 64 f32 features; x tile staged in LDS via
// CDNA5 async global->LDS loads (GLOBAL_LOAD_ASYNC_TO_LDS_B128, ASYNCcnt).
__global__ __launch_bounds__(AM_NODES) void k_argmax(const float* __restrict__ x,
                                                     long long* __restrict__ colors,
                                                     long long* __restrict__ out, int n) {
  __shared__ float lds[AM_NODES * AM_PAD];   // 128 * 68 * 4 = 34816 B
  const int t    = threadIdx.x;
  const int base = blockIdx.x * AM_NODES;

  // 128 rows x 16 chunks(16B) ; 128 threads -> 16 async rounds per thread
  for (int c = t; c < AM_NODES * 16; c += AM_NODES) {
    const int row = c >> 4, sub = c & 15;
    const int node = base + row;
    if (node < n) {
      const float* g = x + (size_t)node * N_FEAT + sub * 4;
      unsigned int laddr = lds_addr32(&lds[row * AM_PAD + sub * 4]);
      asm volatile("global_load_async_to_lds_b128 %0, %1, off"
                   :: "v"(laddr), "v"(g) : "memory");
    }
  }
  async_wait0();
  __syncthreads();

  const int node = base + t;
  if (node < n) {
    const float* row = &lds[t * AM_PAD];
    float best = row[0];
    int   bi   = 0;
#pragma unroll
    for (int j = 1; j < N_FEAT; ++j) {
      float v = row[j];
      if (v > best) { best = v; bi = j; }   // strict '>' => first max (matches jnp.argmax)
    }
    colors[node]       = (long long)bi;
    out[node * 4 + 0]  = (long long)bi;
  }
}

__global__ void k_init_offset(long long* off) { if (threadIdx.x == 0 && blockIdx.x == 0) off[0] = 0; }

// h = splitmix64(colors + 1); nb_sum = 0
__global__ void k_hash_zero(const long long* __restrict__ colors,
                            unsigned long long* __restrict__ h,
                            unsigned long long* __restrict__ nb, int n) {
  int i = blockIdx.x * blockDim.x + threadIdx.x;
  if (i >= n) return;
  h[i]  = splitmix64((unsigned long long)colors[i] + 1ull);
  nb[i] = 0ull;
}

__global__ void k_clear_table(unsigned long long* __restrict__ tkey,
                              unsigned int* __restrict__ tmin) {
  unsigned int s = blockIdx.x * blockDim.x + threadIdx.x;
  if (s >= HASH_M) return;
  tkey[s] = EMPTY64;
  tmin[s] = 0xFFFFFFFFu;
}

// nb_sum[dst] += h[src] over all edges (u64 L2 atomics, order-invariant)
__global__ void k_scatter(const long long* __restrict__ src,
                          const long long* __restrict__ dst,
                          const unsigned long long* __restrict__ h,
                          unsigned long long* __restrict__ nb, int e) {
  int i = blockIdx.x * blockDim.x + threadIdx.x;
  if (i >= e) return;
  if (((i & 15) == 0) && (i + 8192 < e)) {  // one prefetch per 128B line -> global_prefetch_b8
    __builtin_prefetch(&src[i + 8192], 0, 1);
    __builtin_prefetch(&dst[i + 8192], 0, 1);
  }
  const int s = (int)src[i];
  const int d = (int)dst[i];
  atomicAdd(&nb[d], h[s]);
}

// sig = splitmix64(h*FNV + nb); insert into hash table; record min node index
__global__ void k_sig_insert(const unsigned long long* __restrict__ h,
                             const unsigned long long* __restrict__ nb,
                             unsigned long long* __restrict__ tkey,
                             unsigned int* __restrict__ tmin,
                             unsigned int* __restrict__ slot, int n) {
  int i = blockIdx.x * blockDim.x + threadIdx.x;
  if (i >= n) return;
  const unsigned long long sig = splitmix64(h[i] * 0x100000001B3ull + nb[i]);
  unsigned int pos = (unsigned int)sig & (HASH_M - 1u);
  for (;;) {
    unsigned long long prev = atomicCAS(&tkey[pos], EMPTY64, sig);
    if (prev == EMPTY64 || prev == sig) break;
    pos = (pos + 1u) & (HASH_M - 1u);
  }
  atomicMin(&tmin[pos], (unsigned int)i);
  slot[i] = pos;
}

// flag first-occurrence nodes
__global__ void k_flag(const unsigned int* __restrict__ slot,
                       const unsigned int* __restrict__ tmin,
                       unsigned int* __restrict__ prefix, int n) {
  int i = blockIdx.x * blockDim.x + threadIdx.x;
  if (i >= n) return;
  prefix[i] = (tmin[slot[i]] == (unsigned int)i) ? 1u : 0u;
}

// ---- WMMA block scan: 256 flags per wave32 as a 16x16 f32 matrix ----
// incl = V*U + L*RS  (U = upper-tri ones incl diag, L = strictly-lower ones,
// RS[k][n] = rowsum(k)).  Each 16x16x16 product = 4 chained K=4 f32 WMMAs.
// All values are integers <= 256 -> exact in f32.
__global__ __launch_bounds__(32) void k_scan_block(unsigned int* __restrict__ prefix,
                                                   unsigned int* __restrict__ bsum, int n) {
  __shared__ float vv[16 * 17];              // padded rows: bank-conflict-free gathers
  const int  l    = threadIdx.x;             // 0..31 (wave32)
  const int  ln   = l & 15;
  const bool hi   = (l >= 16);
  const int  base = blockIdx.x * 256;

  // load flags in the C/D-matrix lane layout; stash to LDS for A-operand gathers
  float v[8];
#pragma unroll
  for (int g = 0; g < 8; ++g) {
    const int row = hi ? (g + 8) : g;
    const int e   = base + row * 16 + ln;
    float x = (e < n) ? (float)prefix[e] : 0.0f;
    v[g] = x;
    vv[row * 17 + ln] = x;
  }
  __syncthreads();

  // P = V x U : row-wise inclusive prefix
  v8f acc = {0.f, 0.f, 0.f, 0.f, 0.f, 0.f, 0.f, 0.f};
#pragma unroll
  for (int k = 0; k < 4; ++k) {
    const int c0 = 4 * k + (hi ? 2 : 0);     // A: K=0,1 lanes 0-15 ; K=2,3 lanes 16-31
    v2f a, b;
    a.x = vv[ln * 17 + c0];
    a.y = vv[ln * 17 + c0 + 1];
    b.x = (c0     <= ln) ? 1.0f : 0.0f;      // U rows c0, c0+1 at column ln
    b.y = (c0 + 1 <= ln) ? 1.0f : 0.0f;
    acc = __builtin_amdgcn_wmma_f32_16x16x4_f32(false, a, false, b, (short)0, acc,
                                                false, false);
  }

  // row sums rs(m) = P[m][15] : m<8 -> vgpr m lane 15 ; m>=8 -> vgpr m-8 lane 31
  float rs[16];
#pragma unroll
  for (int m = 0; m < 16; ++m)
    rs[m] = __shfl(acc[m & 7], (m < 8) ? 15 : 31);

  // acc += L x RS : add sum of all previous rows (C operand carries P)
#pragma unroll
  for (int k = 0; k < 4; ++k) {
    const int c0 = 4 * k + (hi ? 2 : 0);
    v2f a, b;
    a.x = (c0     < ln) ? 1.0f : 0.0f;       // L[m][c0]  (strictly lower)
    a.y = (c0 + 1 < ln) ? 1.0f : 0.0f;
    b.x = hi ? rs[4 * k + 2] : rs[4 * k + 0];
    b.y = hi ? rs[4 * k + 3] : rs[4 * k + 1];
    acc = __builtin_amdgcn_wmma_f32_16x16x4_f32(false, a, false, b, (short)0, acc,
                                                false, false);
  }

  // write exclusive prefix + block total
#pragma unroll
  for (int g = 0; g < 8; ++g) {
    const int row = hi ? (g + 8) : g;
    const int e   = base + row * 16 + ln;
    if (e < n) prefix[e] = (unsigned int)(acc[g] - v[g] + 0.5f);
  }
  if (l == 31) bsum[blockIdx.x] = (unsigned int)(acc[7] + 0.5f);  // Z[15][15]
}

__global__ void k_scan_top(const unsigned int* __restrict__ bsum,
                           unsigned int* __restrict__ boff,
                           unsigned int* __restrict__ total, int nblk) {
  __shared__ unsigned int tmp[512];
  const int t = threadIdx.x;
  unsigned int v = (t < nblk) ? bsum[t] : 0u;
  tmp[t] = v;
  __syncthreads();
  for (int off = 1; off < 512; off <<= 1) {
    unsigned int a = (t >= off) ? tmp[t - off] : 0u;
    __syncthreads();
    tmp[t] += a;
    __syncthreads();
  }
  if (t < nblk) boff[t] = tmp[t] - v;             // exclusive block offsets
  if (t == 511) total[0] = tmp[511];              // num_new (total uniques)
}

__global__ void k_scan_add(unsigned int* __restrict__ prefix,
                           const unsigned int* __restrict__ boff, int n) {
  int gid = blockIdx.x * 256 + threadIdx.x;
  if (gid < n) prefix[gid] += boff[blockIdx.x];
}

// new color = offset + rank(first index of my signature); write output column
__global__ void k_newcolor(const unsigned int* __restrict__ slot,
                           const unsigned int* __restrict__ tmin,
                           const unsigned int* __restrict__ prefix,
                           const long long* __restrict__ off,
                           long long* __restrict__ colors,
                           long long* __restrict__ out, int n, int col) {
  int i = blockIdx.x * blockDim.x + threadIdx.x;
  if (i >= n) return;
  const unsigned int fi = tmin[slot[i]];
  const long long c = off[0] + (long long)prefix[fi];
  colors[i]         = c;
  out[i * 4 + col]  = c;
}

__global__ void k_offset_update(long long* off, const unsigned int* total) {
  if (threadIdx.x == 0 && blockIdx.x == 0) off[0] += (long long)total[0];
}

// ---------------- launcher ----------------
extern "C" void kernel_launch(void* const* d_in, const int* in_sizes, int n_in,
                              void* d_out, int out_size, void* d_ws, size_t ws_size,
                              hipStream_t stream) {
  (void)in_sizes; (void)n_in; (void)out_size; (void)ws_size;

  const float*     x   = (const float*)d_in[0];
  const long long* ei  = (const long long*)d_in[1];   // int64 [2, E]
  const long long* src = ei;
  const long long* dst = ei + N_EDGES;
  long long*       out = (long long*)d_out;           // int64 [N, 4]

  char* w = (char*)d_ws;
  auto carve = [&](size_t bytes) -> char* {
    char* p = w;
    w += (bytes + 255) & ~(size_t)255;
    return p;
  };
  long long*          colors = (long long*)         carve((size_t)N_NODES * 8);
  unsigned long long* h      = (unsigned long long*)carve((size_t)N_NODES * 8);
  unsigned long long* nb     = (unsigned long long*)carve((size_t)N_NODES * 8);
  unsigned int*       slot   = (unsigned int*)      carve((size_t)N_NODES * 4);
  unsigned int*       prefix = (unsigned int*)      carve((size_t)N_NODES * 4);
  unsigned long long* tkey   = (unsigned long long*)carve((size_t)HASH_M * 8);
  unsigned int*       tmin   = (unsigned int*)      carve((size_t)HASH_M * 4);
  unsigned int*       bsum   = (unsigned int*)      carve(512 * 4);
  unsigned int*       boff   = (unsigned int*)      carve(512 * 4);
  unsigned int*       total  = (unsigned int*)      carve(256);
  long long*          offset = (long long*)         carve(256);

  const int NBn = (N_NODES + 255) / 256;              // 391 node blocks
  const int NBm = (HASH_M + 255) / 256;
  const int NBe = (N_EDGES + 255) / 256;
  const int NBa = (N_NODES + AM_NODES - 1) / AM_NODES;

  k_init_offset<<<1, 1, 0, stream>>>(offset);
  k_argmax<<<NBa, AM_NODES, 0, stream>>>(x, colors, out, N_NODES);

  for (int it = 0; it < NUM_ITERS; ++it) {
    k_hash_zero  <<<NBn, 256, 0, stream>>>(colors, h, nb, N_NODES);
    k_clear_table<<<NBm, 256, 0, stream>>>(tkey, tmin);
    k_scatter    <<<NBe, 256, 0, stream>>>(src, dst, h, nb, N_EDGES);
    k_sig_insert <<<NBn, 256, 0, stream>>>(h, nb, tkey, tmin, slot, N_NODES);
    k_flag       <<<NBn, 256, 0, stream>>>(slot, tmin, prefix, N_NODES);
    k_scan_block <<<NBn, 32,  0, stream>>>(prefix, bsum, N_NODES);
    k_scan_top   <<<1,   512, 0, stream>>>(bsum, boff, total, NBn);
    k_scan_add   <<<NBn, 256, 0, stream>>>(prefix, boff, N_NODES);
    k_newcolor   <<<NBn, 256, 0, stream>>>(slot, tmin, prefix, offset, colors, out, N_NODES, it + 1);
    k_offset_update<<<1, 1, 0, stream>>>(offset, total);
  }
}